// AttentiveM3GNetBlock_35605278884025
// MI455X (gfx1250) — compile-verified
//
#include <hip/hip_runtime.h>
#include <hip/hip_bf16.h>

typedef __attribute__((ext_vector_type(2))) float v2f;
typedef __attribute__((ext_vector_type(8))) float v8f;

#define TW 4  // waves per block for GEMM kernels (wave32)

__device__ __forceinline__ float silu(float x) {
    return x / (1.0f + __expf(-x));
}

// order-preserving float<->uint mapping for atomic max
__device__ __forceinline__ unsigned ford(float f) {
    unsigned u = __float_as_uint(f);
    return (u >> 31) ? ~u : (u | 0x80000000u);
}
__device__ __forceinline__ float forddec(unsigned u) {
    return (u >> 31) ? __uint_as_float(u & 0x7FFFFFFFu) : __uint_as_float(~u);
}

// Cooperatively load a [rows x COLS] f32 weight into LDS in paired-K layout:
//   W[k][c]  ->  sWp[((k>>1)*COLS + c)*2 + (k&1)]
// so a WMMA B fragment (k0, k0+1 at column c) is one aligned 8-byte LDS load.
template <int COLS, int TOTAL>
__device__ __forceinline__ void load_w_paired(const float* __restrict__ W,
                                              float* __restrict__ sWp, int tid) {
    #pragma unroll
    for (int i4 = tid * 4; i4 < TOTAL; i4 += 128 * 4) {
        float4 v = *(const float4*)(W + i4);
        const int k = i4 / COLS;
        const int c = i4 - k * COLS;
        float* d = sWp + ((((k >> 1) * COLS) + c) << 1) + (k & 1);
        d[0] = v.x; d[2] = v.y; d[4] = v.z; d[6] = v.w;
    }
}

// ---------------------------------------------------------------------------
// Edge MLP: eo = silu(silu(ef @ W1 + b1) @ W2 + b2)
// ef: [E,64], W1: [64,128], W2: [128,64]. One wave per 16-edge tile.
// ---------------------------------------------------------------------------
__global__ __launch_bounds__(128) void edge_mlp_kernel(
    const float* __restrict__ ef,
    const float* __restrict__ W1, const float* __restrict__ b1,
    const float* __restrict__ W2, const float* __restrict__ b2,
    float* __restrict__ eo, int nTiles)
{
    __shared__ float sW[64 * 128];       // 32KB: paired W1, later paired W2
    __shared__ float sT[TW][16 * 128];   // 32KB: per-wave A tile / hidden / out tile

    const int tid  = threadIdx.x;
    const int w    = tid >> 5;
    const int lane = tid & 31;
    const int hw   = lane >> 4;   // lane half
    const int lm   = lane & 15;

    int tile0 = blockIdx.x * TW + w;
    bool active = tile0 < nTiles;
    int tile = active ? tile0 : (nTiles - 1);

    load_w_paired<128, 64 * 128>(W1, sW, tid);
    // stage A tile 16x64 (1024 floats per wave, float4)
    {
        const float4* s4 = (const float4*)(ef + (size_t)tile * (16 * 64));
        float4* d4 = (float4*)&sT[w][0];
        #pragma unroll
        for (int i = lane; i < 256; i += 32) d4[i] = s4[i];
    }
    __syncthreads();

    // GEMM1: 16x64 @ 64x128 -> acc[8] column tiles of 16
    v8f acc[8];
    #pragma unroll
    for (int n = 0; n < 8; ++n) acc[n] = (v8f)0.0f;
    #pragma unroll
    for (int kb = 0; kb < 16; ++kb) {
        const int k0 = kb * 4 + 2 * hw;
        v2f a = *(const v2f*)&sT[w][lm * 64 + k0];
        #pragma unroll
        for (int n = 0; n < 8; ++n) {
            v2f b = *(const v2f*)&sW[(((kb * 2 + hw) * 128) + n * 16 + lm) << 1];
            acc[n] = __builtin_amdgcn_wmma_f32_16x16x4_f32(
                false, a, false, b, (short)0, acc[n], false, false);
        }
    }
    __syncthreads();  // all waves done reading sW (W1) before overwrite

    // hidden = silu(acc + b1) -> sT[w] row-major 16x128
    #pragma unroll
    for (int n = 0; n < 8; ++n) {
        float bb = b1[n * 16 + lm];
        #pragma unroll
        for (int r = 0; r < 8; ++r) {
            sT[w][(r + 8 * hw) * 128 + n * 16 + lm] = silu(acc[n][r] + bb);
        }
    }
    load_w_paired<64, 128 * 64>(W2, sW, tid);
    __syncthreads();

    // GEMM2: 16x128 @ 128x64 -> acc2[4]
    v8f acc2[4];
    #pragma unroll
    for (int n = 0; n < 4; ++n) acc2[n] = (v8f)0.0f;
    #pragma unroll
    for (int kb = 0; kb < 32; ++kb) {
        const int k0 = kb * 4 + 2 * hw;
        v2f a = *(const v2f*)&sT[w][lm * 128 + k0];
        #pragma unroll
        for (int n = 0; n < 4; ++n) {
            v2f b = *(const v2f*)&sW[(((kb * 2 + hw) * 64) + n * 16 + lm) << 1];
            acc2[n] = __builtin_amdgcn_wmma_f32_16x16x4_f32(
                false, a, false, b, (short)0, acc2[n], false, false);
        }
    }
    // restage out tile 16x64 through LDS for coalesced b128 stores
    #pragma unroll
    for (int n = 0; n < 4; ++n) {
        float bb = b2[n * 16 + lm];
        #pragma unroll
        for (int r = 0; r < 8; ++r) {
            sT[w][(r + 8 * hw) * 64 + n * 16 + lm] = silu(acc2[n][r] + bb);
        }
    }
    __syncthreads();
    if (active) {
        const float4* s4 = (const float4*)&sT[w][0];
        float4* o4 = (float4*)(eo + (size_t)tile * (16 * 64));
        #pragma unroll
        for (int j = 0; j < 8; ++j) o4[lane + j * 32] = s4[lane + j * 32];
    }
}

// ---------------------------------------------------------------------------
// QKV projection: out = nf @ W  for W in {Wq,Wk,Wv} (blockIdx.y selects)
// nf: [N,128], W: [128,128]. Weight loaded in two paired 64-row halves.
// ---------------------------------------------------------------------------
__global__ __launch_bounds__(128) void qkv_kernel(
    const float* __restrict__ nf,
    const float* __restrict__ Wq, const float* __restrict__ Wk,
    const float* __restrict__ Wv,
    float* __restrict__ Qo, float* __restrict__ Ko, float* __restrict__ Vo,
    int nTiles)
{
    __shared__ float sW[64 * 128];
    __shared__ float sT[TW][16 * 128];

    const int tid  = threadIdx.x;
    const int w    = tid >> 5;
    const int lane = tid & 31;
    const int hw   = lane >> 4;
    const int lm   = lane & 15;

    const float* W = (blockIdx.y == 0) ? Wq : ((blockIdx.y == 1) ? Wk : Wv);
    float* O       = (blockIdx.y == 0) ? Qo : ((blockIdx.y == 1) ? Ko : Vo);

    int tile0 = blockIdx.x * TW + w;
    bool active = tile0 < nTiles;
    int tile = active ? tile0 : (nTiles - 1);

    // stage A tile 16x128 per wave
    {
        const float4* s4 = (const float4*)(nf + (size_t)tile * (16 * 128));
        float4* d4 = (float4*)&sT[w][0];
        #pragma unroll
        for (int i = lane; i < 512; i += 32) d4[i] = s4[i];
    }
    load_w_paired<128, 64 * 128>(W, sW, tid);  // K rows 0..63
    __syncthreads();

    v8f acc[8];
    #pragma unroll
    for (int n = 0; n < 8; ++n) acc[n] = (v8f)0.0f;
    #pragma unroll
    for (int kb = 0; kb < 16; ++kb) {
        const int k0 = kb * 4 + 2 * hw;
        v2f a = *(const v2f*)&sT[w][lm * 128 + k0];
        #pragma unroll
        for (int n = 0; n < 8; ++n) {
            v2f b = *(const v2f*)&sW[(((kb * 2 + hw) * 128) + n * 16 + lm) << 1];
            acc[n] = __builtin_amdgcn_wmma_f32_16x16x4_f32(
                false, a, false, b, (short)0, acc[n], false, false);
        }
    }
    __syncthreads();
    load_w_paired<128, 64 * 128>(W + 64 * 128, sW, tid);  // K rows 64..127
    __syncthreads();
    #pragma unroll
    for (int kb = 0; kb < 16; ++kb) {
        const int k0 = kb * 4 + 2 * hw;
        v2f a = *(const v2f*)&sT[w][lm * 128 + 64 + k0];
        #pragma unroll
        for (int n = 0; n < 8; ++n) {
            v2f b = *(const v2f*)&sW[(((kb * 2 + hw) * 128) + n * 16 + lm) << 1];
            acc[n] = __builtin_amdgcn_wmma_f32_16x16x4_f32(
                false, a, false, b, (short)0, acc[n], false, false);
        }
    }
    __syncthreads();  // all waves done with own sT (A) region before restage
    // restage out tile 16x128 through LDS for coalesced b128 stores
    #pragma unroll
    for (int n = 0; n < 8; ++n) {
        #pragma unroll
        for (int r = 0; r < 8; ++r) {
            sT[w][(r + 8 * hw) * 128 + n * 16 + lm] = acc[n][r];
        }
    }
    __syncthreads();
    if (active) {
        const float4* s4 = (const float4*)&sT[w][0];
        float4* o4 = (float4*)(O + (size_t)tile * (16 * 128));
        #pragma unroll
        for (int j = 0; j < 16; ++j) o4[lane + j * 32] = s4[lane + j * 32];
    }
}

// ---------------------------------------------------------------------------
// scores[e] = dot(Q[dst[e]], K[src[e]]) / sqrt(128)
// ---------------------------------------------------------------------------
__global__ __launch_bounds__(256) void score_kernel(
    const float* __restrict__ Q, const float* __restrict__ Km,
    const int* __restrict__ src, const int* __restrict__ dst,
    float* __restrict__ sc, int E)
{
    int e = blockIdx.x * 256 + threadIdx.x;
    if (e >= E) return;
    const float4* q = (const float4*)(Q + (size_t)dst[e] * 128);
    const float4* k = (const float4*)(Km + (size_t)src[e] * 128);
    float acc = 0.0f;
    #pragma unroll
    for (int i = 0; i < 32; ++i) {
        float4 a = q[i], b = k[i];
        acc += a.x * b.x + a.y * b.y + a.z * b.z + a.w * b.w;
    }
    sc[e] = acc * 0.08838834764831845f;  // 1/sqrt(128)
}

__global__ __launch_bounds__(256) void segmax_kernel(
    const float* __restrict__ sc, const int* __restrict__ dst,
    unsigned* __restrict__ mx, int E)
{
    int e = blockIdx.x * 256 + threadIdx.x;
    if (e >= E) return;
    atomicMax(&mx[dst[e]], ford(sc[e]));
}

__global__ __launch_bounds__(256) void expsum_kernel(
    float* __restrict__ sc, const int* __restrict__ dst,
    const unsigned* __restrict__ mx, float* __restrict__ denom, int E)
{
    int e = blockIdx.x * 256 + threadIdx.x;
    if (e >= E) return;
    int d = dst[e];
    float v = __expf(sc[e] - forddec(mx[d]));
    sc[e] = v;
    atomicAdd(&denom[d], v);
}

// one wave per edge: lane handles 4 columns (float4 of V row)
__global__ __launch_bounds__(256) void scatter_kernel(
    const float* __restrict__ V, const float* __restrict__ sc,
    const float* __restrict__ denom,
    const int* __restrict__ src, const int* __restrict__ dst,
    float* __restrict__ nodeOut, int E)
{
    int gw = (blockIdx.x * 256 + threadIdx.x) >> 5;
    int lane = threadIdx.x & 31;
    if (gw >= E) return;
    int s = src[gw], d = dst[gw];
    float alpha = sc[gw] / denom[d];
    float4 v = ((const float4*)(V + (size_t)s * 128))[lane];
    float* out = nodeOut + (size_t)d * 128 + lane * 4;
    atomicAdd(out + 0, alpha * v.x);
    atomicAdd(out + 1, alpha * v.y);
    atomicAdd(out + 2, alpha * v.z);
    atomicAdd(out + 3, alpha * v.w);
}

__global__ __launch_bounds__(256) void init_kernel(
    float* __restrict__ nodeOut, unsigned* __restrict__ mx,
    float* __restrict__ denom, int nNodes, int nOut)
{
    int i = blockIdx.x * 256 + threadIdx.x;
    if (i < nOut) nodeOut[i] = 0.0f;
    if (i < nNodes) { mx[i] = 0u; denom[i] = 0.0f; }
}

// ---------------------------------------------------------------------------
extern "C" void kernel_launch(void* const* d_in, const int* in_sizes, int n_in,
                              void* d_out, int out_size, void* d_ws, size_t ws_size,
                              hipStream_t stream)
{
    const float* edge_feat = (const float*)d_in[0];
    const float* node_feat = (const float*)d_in[1];
    const int*   src       = (const int*)d_in[2];
    const int*   dst       = (const int*)d_in[3];
    const float* Wq        = (const float*)d_in[4];
    const float* Wk        = (const float*)d_in[5];
    const float* Wv        = (const float*)d_in[6];
    const float* W1        = (const float*)d_in[7];
    const float* b1        = (const float*)d_in[8];
    const float* W2        = (const float*)d_in[9];
    const float* b2        = (const float*)d_in[10];

    const int E = in_sizes[2];            // 800000
    const int N = in_sizes[1] / 128;      // 50000

    float* edge_out = (float*)d_out;                       // [E,64]
    float* node_out = (float*)d_out + (size_t)E * 64;      // [N,128]

    // workspace layout
    float* Qb = (float*)d_ws;
    float* Kb = Qb + (size_t)N * 128;
    float* Vb = Kb + (size_t)N * 128;
    float* sc = Vb + (size_t)N * 128;
    unsigned* mx = (unsigned*)(sc + E);
    float* denom = (float*)(mx + N);

    const int nOut = N * 128;
    init_kernel<<<(nOut + 255) / 256, 256, 0, stream>>>(node_out, mx, denom, N, nOut);

    const int eTiles = E / 16;                 // 50000 (exact)
    edge_mlp_kernel<<<(eTiles + TW - 1) / TW, 128, 0, stream>>>(
        edge_feat, W1, b1, W2, b2, edge_out, eTiles);

    const int nTiles = N / 16;                 // 3125 (exact)
    qkv_kernel<<<dim3((nTiles + TW - 1) / TW, 3), 128, 0, stream>>>(
        node_feat, Wq, Wk, Wv, Qb, Kb, Vb, nTiles);

    score_kernel<<<(E + 255) / 256, 256, 0, stream>>>(Qb, Kb, src, dst, sc, E);
    segmax_kernel<<<(E + 255) / 256, 256, 0, stream>>>(sc, dst, mx, E);
    expsum_kernel<<<(E + 255) / 256, 256, 0, stream>>>(sc, dst, mx, denom, E);

    const long long sThreads = (long long)E * 32;
    scatter_kernel<<<(int)((sThreads + 255) / 256), 256, 0, stream>>>(
        Vb, sc, denom, src, dst, node_out, E);
}